// GATLayer_48979807043936
// MI455X (gfx1250) — compile-verified
//
#include <hip/hip_runtime.h>

typedef float v2f __attribute__((ext_vector_type(2)));
typedef float v8f __attribute__((ext_vector_type(8)));

#define IN_DIM   128
#define D_DIM    64
#define K_NB     32
#define OUT_COLS 95   // (K-1) + D = 31 + 64
#define BN_EPS   1e-5f
#define NEG_SLOPE 0.01f

// ---------------------------------------------------------------------------
// Kernel 1: z[N,64] = h[N,128] @ W_fc[64,128]^T  via fp32 WMMA 16x16x4.
// One block = 128 threads = 4 waves; wave w computes z cols [16w,16w+16) for
// the block's 16-row strip. 32 chained v_wmma_f32_16x16x4_f32 per wave.
// A layout (32b, 16x4): lanes 0-15 -> K={k0,k0+1}, lanes 16-31 -> K={k0+2,k0+3}
// B layout mirrored; C/D: vgpr v, lane l -> M = v + 8*(l>=16), Nc = l&15.
// ---------------------------------------------------------------------------
__global__ __launch_bounds__(128) void zgemm_kernel(
    const float* __restrict__ h, const float* __restrict__ Wfc,
    float* __restrict__ z) {
  const int wave = threadIdx.x >> 5;
  const int lane = threadIdx.x & 31;
  const int half = lane >> 4;
  const int l16  = lane & 15;
  const int row0 = blockIdx.x << 4;
  const int col0 = wave << 4;
  const float* __restrict__ pa = h   + (size_t)(row0 + l16) * IN_DIM + 2 * half;
  const float* __restrict__ pb = Wfc + (size_t)(col0 + l16) * IN_DIM + 2 * half;
  v8f acc = {};
#pragma unroll
  for (int k0 = 0; k0 < IN_DIM; k0 += 4) {
    v2f a, b;
    a.x = pa[k0]; a.y = pa[k0 + 1];
    b.x = pb[k0]; b.y = pb[k0 + 1];
    acc = __builtin_amdgcn_wmma_f32_16x16x4_f32(
        /*neg_a=*/false, a, /*neg_b=*/false, b,
        /*c_mod=*/(short)0, acc, /*reuse_a=*/false, /*reuse_b=*/false);
  }
  float* zp = z + (size_t)(row0 + 8 * half) * D_DIM + col0 + l16;
#pragma unroll
  for (int v = 0; v < 8; ++v) zp[(size_t)v * D_DIM] = acc[v];
}

// ---------------------------------------------------------------------------
// Kernel 2: e_src[n] = z[n,:] . a[:64];  e_dst[n] = z[n,:] . a[64:128]
// ---------------------------------------------------------------------------
__global__ __launch_bounds__(256) void attn_e_kernel(
    const float* __restrict__ z, const float* __restrict__ a_attn,
    float* __restrict__ e_src, float* __restrict__ e_dst, int N) {
  int n = blockIdx.x * 256 + threadIdx.x;
  if (n >= N) return;
  const float4* zr = (const float4*)(z + (size_t)n * D_DIM);
  float s = 0.f, d = 0.f;
#pragma unroll
  for (int i = 0; i < 16; ++i) {
    float4 v = zr[i];
    s = fmaf(v.x, a_attn[4*i+0], s); s = fmaf(v.y, a_attn[4*i+1], s);
    s = fmaf(v.z, a_attn[4*i+2], s); s = fmaf(v.w, a_attn[4*i+3], s);
    d = fmaf(v.x, a_attn[64+4*i+0], d); d = fmaf(v.y, a_attn[64+4*i+1], d);
    d = fmaf(v.z, a_attn[64+4*i+2], d); d = fmaf(v.w, a_attn[64+4*i+3], d);
  }
  e_src[n] = s; e_dst[n] = d;
}

// ---------------------------------------------------------------------------
// Kernel 3: per-node softmax, gather, alpha*z, row/col convs, pre-BN store
// into d_out, and per-block partial sums for the global BatchNorm stats.
// 256 threads = 4 nodes/block, 64 threads (=2 waves) per node (thread = d).
// Gathered z-row loads are 256B contiguous per neighbor -> fully coalesced
// L2 hits (z is 16MB << 192MB L2). Deterministic: partials per block, no
// float atomics.
// ---------------------------------------------------------------------------
__global__ __launch_bounds__(256) void gat_main_kernel(
    const float* __restrict__ z, const float* __restrict__ e_src,
    const float* __restrict__ e_dst, const int* __restrict__ src_idx,
    const float* __restrict__ w_row, const float* __restrict__ b_row,
    const float* __restrict__ w_col, const float* __restrict__ b_col,
    float* __restrict__ out, float* __restrict__ partials) {
  __shared__ float feats[4][K_NB][D_DIM + 1];  // +1 pad: conflict-free col reads
  __shared__ float alpha_s[4][K_NB];
  __shared__ int   src_s[4][K_NB];
  __shared__ float r0[256], r1[256], r2[256], r3[256];

  const int tid = threadIdx.x;
  const int g   = tid >> 6;   // node slot 0..3
  const int l   = tid & 63;   // lane within node group
  const int n   = blockIdx.x * 4 + g;

  // ---- Phase A: leaky-relu + softmax over K=32 neighbors (first wave of group)
  if (l < K_NB) {
    const int k = l;
    const int s = src_idx[n * K_NB + k];
    src_s[g][k] = s;
    float e = e_src[s] + e_dst[n];
    e = (e > 0.f) ? e : NEG_SLOPE * e;
    float m = e;
    for (int off = 16; off > 0; off >>= 1) m = fmaxf(m, __shfl_xor(m, off, 32));
    const float ex = __expf(e - m);
    float sum = ex;
    for (int off = 16; off > 0; off >>= 1) sum += __shfl_xor(sum, off, 32);
    alpha_s[g][k] = ex / sum;
  }
  __syncthreads();

  // ---- Phase B: feats + col conv (thread l == dimension d)
  const int d = l;
  float col_acc = 0.f;
#pragma unroll 4
  for (int k = 0; k < K_NB; ++k) {
    const float f = alpha_s[g][k] * z[(size_t)src_s[g][k] * D_DIM + d];
    feats[g][k][d] = f;
    col_acc = fmaf(f, w_col[k * D_DIM + d], col_acc);
  }
  const float colv = col_acc + b_col[0];
  out[(size_t)n * OUT_COLS + (K_NB - 1) + d] = colv;
  __syncthreads();

  // ---- Phase C: row conv (threads l < 31 handle row position k)
  float rowv = 0.f;
  if (l < K_NB - 1) {
    const int k = l;
    float acc = 0.f;
#pragma unroll 8
    for (int dd = 0; dd < D_DIM; ++dd) {
      acc = fmaf(feats[g][k][dd],     w_row[dd],          acc);
      acc = fmaf(feats[g][k + 1][dd], w_row[D_DIM + dd],  acc);
    }
    rowv = acc + b_row[0];
    out[(size_t)n * OUT_COLS + k] = rowv;
  }

  // ---- Phase D: block partial sums for BN (deterministic tree reduce)
  r0[tid] = rowv;          r1[tid] = rowv * rowv;
  r2[tid] = colv;          r3[tid] = colv * colv;
  __syncthreads();
  for (int s = 128; s > 0; s >>= 1) {
    if (tid < s) {
      r0[tid] += r0[tid + s]; r1[tid] += r1[tid + s];
      r2[tid] += r2[tid + s]; r3[tid] += r3[tid + s];
    }
    __syncthreads();
  }
  if (tid == 0) {
    float* p = partials + (size_t)blockIdx.x * 4;
    p[0] = r0[0]; p[1] = r1[0]; p[2] = r2[0]; p[3] = r3[0];
  }
}

// ---------------------------------------------------------------------------
// Kernel 4: single-block reduction of partials -> BN constants.
// stats = {mean_row, scale_row, beta_row, mean_col, scale_col, beta_col}
// ---------------------------------------------------------------------------
__global__ __launch_bounds__(256) void reduce_stats_kernel(
    const float* __restrict__ partials, int nblocks,
    const float* __restrict__ g_row, const float* __restrict__ beta_row,
    const float* __restrict__ g_col, const float* __restrict__ beta_col,
    float* __restrict__ stats, int N) {
  __shared__ float r0[256], r1[256], r2[256], r3[256];
  const int tid = threadIdx.x;
  float a0 = 0.f, a1 = 0.f, a2 = 0.f, a3 = 0.f;
  for (int i = tid; i < nblocks; i += 256) {
    const float* p = partials + (size_t)i * 4;
    a0 += p[0]; a1 += p[1]; a2 += p[2]; a3 += p[3];
  }
  r0[tid] = a0; r1[tid] = a1; r2[tid] = a2; r3[tid] = a3;
  __syncthreads();
  for (int s = 128; s > 0; s >>= 1) {
    if (tid < s) {
      r0[tid] += r0[tid + s]; r1[tid] += r1[tid + s];
      r2[tid] += r2[tid + s]; r3[tid] += r3[tid + s];
    }
    __syncthreads();
  }
  if (tid == 0) {
    const float cr = (float)N * (float)(K_NB - 1);
    const float cc = (float)N * (float)D_DIM;
    const float mr = r0[0] / cr;
    const float vr = r1[0] / cr - mr * mr;
    const float mc = r2[0] / cc;
    const float vc = r3[0] / cc - mc * mc;
    stats[0] = mr; stats[1] = rsqrtf(vr + BN_EPS) * g_row[0]; stats[2] = beta_row[0];
    stats[3] = mc; stats[4] = rsqrtf(vc + BN_EPS) * g_col[0]; stats[5] = beta_col[0];
  }
}

// ---------------------------------------------------------------------------
// Kernel 5: in-place BN + ReLU over d_out [N,95].
// ---------------------------------------------------------------------------
__global__ __launch_bounds__(256) void bn_relu_kernel(
    float* __restrict__ out, const float* __restrict__ stats, size_t total) {
  const float mr = stats[0], sr = stats[1], br = stats[2];
  const float mc = stats[3], sc = stats[4], bc = stats[5];
  size_t i = (size_t)blockIdx.x * 256 + threadIdx.x;
  const size_t stride = (size_t)gridDim.x * 256;
  for (; i < total; i += stride) {
    const int c = (int)(i % OUT_COLS);
    const float x = out[i];
    const float y = (c < K_NB - 1) ? fmaf(x - mr, sr, br) : fmaf(x - mc, sc, bc);
    out[i] = fmaxf(y, 0.f);
  }
}

// ---------------------------------------------------------------------------
extern "C" void kernel_launch(void* const* d_in, const int* in_sizes, int n_in,
                              void* d_out, int out_size, void* d_ws, size_t ws_size,
                              hipStream_t stream) {
  const float* h        = (const float*)d_in[0];
  const float* W_fc     = (const float*)d_in[1];
  const float* a_attn   = (const float*)d_in[2];
  const float* w_row    = (const float*)d_in[3];
  const float* b_row    = (const float*)d_in[4];
  const float* w_col    = (const float*)d_in[5];
  const float* b_col    = (const float*)d_in[6];
  const float* g_row    = (const float*)d_in[7];
  const float* beta_row = (const float*)d_in[8];
  const float* g_col    = (const float*)d_in[9];
  const float* beta_col = (const float*)d_in[10];
  const int*   src_idx  = (const int*)d_in[11];

  const int N = in_sizes[0] / IN_DIM;   // 65536 (multiple of 16 assumed)
  float* out = (float*)d_out;

  // Workspace layout (floats): z[N*64] | e_src[N] | e_dst[N] | partials[N] | stats[6]
  float* z        = (float*)d_ws;
  float* e_src    = z + (size_t)N * D_DIM;
  float* e_dst    = e_src + N;
  float* partials = e_dst + N;
  float* stats    = partials + N;   // (N/4 blocks)*4 = N floats of partials

  const int gatBlocks = N / 4;

  zgemm_kernel<<<N / 16, 128, 0, stream>>>(h, W_fc, z);
  attn_e_kernel<<<(N + 255) / 256, 256, 0, stream>>>(z, a_attn, e_src, e_dst, N);
  gat_main_kernel<<<gatBlocks, 256, 0, stream>>>(z, e_src, e_dst, src_idx,
                                                 w_row, b_row, w_col, b_col,
                                                 out, partials);
  reduce_stats_kernel<<<1, 256, 0, stream>>>(partials, gatBlocks,
                                             g_row, beta_row, g_col, beta_col,
                                             stats, N);
  const size_t total = (size_t)N * OUT_COLS;
  bn_relu_kernel<<<2048, 256, 0, stream>>>(out, stats, total);
}